// RoIAlignAvg_5669356836452
// MI455X (gfx1250) — compile-verified
//
#include <hip/hip_runtime.h>

typedef __attribute__((ext_vector_type(2))) float v2f;
typedef __attribute__((ext_vector_type(8))) float v8f;

#define AH 7
#define AW 7
#define SCALE_ 0.0625f
#define C_ 256
#define H_ 64
#define W_ 64

__global__ __launch_bounds__(256) void RoIAlignAvg_wmma_kernel(
    const float* __restrict__ feat,   // [B, C, H, W]
    const float* __restrict__ rois,   // [N, 5]
    float* __restrict__ out,          // [N, C, 7, 7]
    int nroi)
{
  __shared__ float Ush[16][16];   // pool*row-lerp weights (rows 7..15 zero)
  __shared__ float Vsh[16][16];   // pool*col-lerp weights
  __shared__ int   Ridx[16];      // clamped feature row indices
  __shared__ int   Qidx[16];      // clamped feature col indices
  __shared__ float Tsh[8][256];   // per-wave D-layout -> A-layout bounce

  const int roi = blockIdx.x;
  if (roi >= nroi) return;
  const int tid = threadIdx.x;

  // ---- zero the 16x16 weight matrices (one element per thread) ----
  {
    int r = tid >> 4, cc = tid & 15;
    Ush[r][cc] = 0.0f;
    Vsh[r][cc] = 0.0f;
  }
  __syncthreads();

  const float* roiP = rois + (size_t)roi * 5;

  // ---- per-RoI separable weights: threads 0..7 rows, 8..15 cols ----
  if (tid < 16) {
    const bool isCol = tid >= 8;
    const int  i     = tid & 7;                     // sample index 0..7
    const float p1 = roiP[isCol ? 1 : 2] * SCALE_;  // x1 or y1
    const float p2 = roiP[isCol ? 3 : 4] * SCALE_;  // x2 or y2
    const float len = fmaxf(p2 - p1 + 1.0f, 0.0f);
    const float bin = len / (float)(isCol ? AW : AH);   // /(aw-1) == /7
    const float t   = p1 + (float)i * bin;
    const int   lim = isCol ? W_ : H_;
    const bool  v   = (t >= 0.0f) && (t < (float)lim);
    const float fl  = floorf(t);
    int lo = (int)fl;
    lo = lo < 0 ? 0 : (lo > lim - 1 ? lim - 1 : lo);
    int hi = (lo + 1 > lim - 1) ? (lim - 1) : (lo + 1);
    const float fr = t - fl;
    // fold validity and the 0.5 pooling factor into the lerp taps
    const float w0 = (v ? (1.0f - fr) : 0.0f) * 0.5f;
    const float w1 = (v ? fr          : 0.0f) * 0.5f;
    if (isCol) {
      Qidx[2 * i] = lo; Qidx[2 * i + 1] = hi;
      if (i <= 6) { Vsh[i][2 * i]     = w0; Vsh[i][2 * i + 1]     = w1; }
      if (i >= 1) { Vsh[i - 1][2 * i] = w0; Vsh[i - 1][2 * i + 1] = w1; }
    } else {
      Ridx[2 * i] = lo; Ridx[2 * i + 1] = hi;
      if (i <= 6) { Ush[i][2 * i]     = w0; Ush[i][2 * i + 1]     = w1; }
      if (i >= 1) { Ush[i - 1][2 * i] = w0; Ush[i - 1][2 * i + 1] = w1; }
    }
  }
  __syncthreads();

  // wave id is uniform within the wave: force it scalar so channel-plane
  // base pointers live in SGPRs and memory ops use the SADDR+voffset form.
  const int wave = __builtin_amdgcn_readfirstlane(tid >> 5);
  const int lane = tid & 31;
  const int l    = lane & 15;
  const int koff = (lane < 16) ? 0 : 2;

  // ---- preload per-RoI WMMA fragments + gather BYTE offsets (u32, zext) ----
  v2f Ua[4], Vb[4];
  unsigned off[8];
  #pragma unroll
  for (int kc = 0; kc < 4; ++kc) {
    Ua[kc].x = Ush[l][4 * kc + koff];        // A-operand of GEMM1 (U)
    Ua[kc].y = Ush[l][4 * kc + koff + 1];
    Vb[kc].x = Vsh[l][4 * kc + koff];        // B-operand of GEMM2 (V^T)
    Vb[kc].y = Vsh[l][4 * kc + koff + 1];
    off[2 * kc]     = (unsigned)(Ridx[4 * kc + koff]     * W_ + Qidx[l]) * 4u;
    off[2 * kc + 1] = (unsigned)(Ridx[4 * kc + koff + 1] * W_ + Qidx[l]) * 4u;
  }

  const int b = __builtin_amdgcn_readfirstlane((int)roiP[0]);
  const char* fbase = (const char*)(feat + (size_t)(b * C_) * (H_ * W_));
  char* obase = (char*)(out + (size_t)roi * C_ * (AH * AW));
  float* tw = &Tsh[wave][0];
  const int prow = (lane < 16) ? 0 : 8;
  const unsigned laneB = (unsigned)lane * 4u;

  // ---- channel loop: 8 gathers + 8 v_wmma_f32_16x16x4_f32 per channel ----
  for (int c = wave; c < C_; c += 8) {
    const char* fp = fbase + (size_t)c * (H_ * W_ * 4);   // SGPR base (c scalar)

    // GEMM1: T(16x16) = U(16x16,K=r) * M(patch, gathered directly into B slots)
    v8f T = {};
    #pragma unroll
    for (int kc = 0; kc < 4; ++kc) {
      v2f Bm;
      Bm.x = *(const float*)(fp + (size_t)off[2 * kc]);
      Bm.y = *(const float*)(fp + (size_t)off[2 * kc + 1]);
      T = __builtin_amdgcn_wmma_f32_16x16x4_f32(
              false, Ua[kc], false, Bm, (short)0, T, false, false);
    }

    // redistribute T from C/D layout to A layout (cross-lane, via wave LDS)
    #pragma unroll
    for (int k = 0; k < 8; ++k)
      tw[(prow + k) * 16 + l] = T[k];
    asm volatile("s_wait_dscnt 0x0" ::: "memory");

    // GEMM2: O = T * V^T  (K = q)
    v8f O = {};
    #pragma unroll
    for (int kc = 0; kc < 4; ++kc) {
      v2f Ta;
      Ta.x = tw[l * 16 + 4 * kc + koff];
      Ta.y = tw[l * 16 + 4 * kc + koff + 1];
      O = __builtin_amdgcn_wmma_f32_16x16x4_f32(
              false, Ta, false, Vb[kc], (short)0, O, false, false);
    }

    // lanes 0..6 hold O[pi=k][pj=lane] for k=0..6; stream out non-temporally
    if (lane < 7) {
      char* op = obase + (size_t)c * (AH * AW * 4) + (size_t)laneB;
      #pragma unroll
      for (int k = 0; k < 7; ++k)
        __builtin_nontemporal_store(O[k], (float*)(op + k * 28));
    }
  }
}

extern "C" void kernel_launch(void* const* d_in, const int* in_sizes, int n_in,
                              void* d_out, int out_size, void* d_ws, size_t ws_size,
                              hipStream_t stream) {
  const float* feat = (const float*)d_in[0];
  const float* rois = (const float*)d_in[1];
  float* out = (float*)d_out;
  const int nroi = in_sizes[1] / 5;
  RoIAlignAvg_wmma_kernel<<<nroi, 256, 0, stream>>>(feat, rois, out, nroi);
}